// Processor_29411936043420
// MI455X (gfx1250) — compile-verified
//
#include <hip/hip_runtime.h>
#include <math.h>

// ---------------------------------------------------------------------------
// 4-layer GAT (PyG GATConv, heads=1) for MI455X / gfx1250, wave32.
//
// Algebraic fusion: (ea @ We) @ a_e == ea @ (We @ a_e), so the huge per-layer
// edge GEMM [650k,128]x[128,128] collapses to a per-edge dot with a
// precomputed 128-vector; edge_attr is read ONCE for all 4 layers.
// Node GEMM x@W stays full fp32 via V_WMMA_F32_16X16X4_F32.
// ---------------------------------------------------------------------------

#define NN   50000
#define EE   600000
#define DD   128
#define LL   4
#define ETOT (EE + NN)
#define NEG_SLOPE 0.2f
#define LN_EPS    1e-5f

typedef float v2f __attribute__((ext_vector_type(2)));
typedef float v8f __attribute__((ext_vector_type(8)));

__device__ __forceinline__ float waveReduceSum(float v) {
#pragma unroll
  for (int off = 16; off >= 1; off >>= 1)
    v += __shfl_xor(v, off, 32);
  return v;
}

__device__ __forceinline__ void atomicMaxF(float* addr, float val) {
  int* ai = (int*)addr;
  int old = *ai;
  while (__int_as_float(old) < val) {
    int assumed = old;
    old = atomicCAS(ai, assumed, __float_as_int(val));
    if (old == assumed) break;
  }
}

// --------------------------- phase 0 kernels -------------------------------

// scatter-mean accumulation for self-loop edge_attr (esum, cnt)
__global__ __launch_bounds__(256) void k_edge_accum(
    const float* __restrict__ ea, const int* __restrict__ dst,
    float* __restrict__ esum, float* __restrict__ cnt) {
  int e = blockIdx.x * 8 + (threadIdx.x >> 5);
  if (e >= EE) return;
  int lane = threadIdx.x & 31;
  int d = dst[e];
  float4 v = ((const float4*)(ea + (size_t)e * DD))[lane];
  float* p = esum + (size_t)d * DD + lane * 4;
  atomicAdd(p + 0, v.x);
  atomicAdd(p + 1, v.y);
  atomicAdd(p + 2, v.z);
  atomicAdd(p + 3, v.w);
  if (lane == 0) atomicAdd(cnt + d, 1.0f);
}

// w_ae[l][j] = dot(We[l][j][:], a_e[l][:])
__global__ __launch_bounds__(128) void k_wae(
    const float* __restrict__ We, const float* __restrict__ ae,
    float* __restrict__ wae) {
  __shared__ float red[DD];
  int l = blockIdx.x >> 7;
  int j = blockIdx.x & 127;
  int k = threadIdx.x;
  red[k] = We[((size_t)l * DD + j) * DD + k] * ae[l * DD + k];
  __syncthreads();
  for (int s = 64; s >= 1; s >>= 1) {
    if (k < s) red[k] += red[k + s];
    __syncthreads();
  }
  if (k == 0) wae[l * DD + j] = red[0];
}

// ealpha[l][e] = dot(edge_attr[e], w_ae[l])   for all 4 layers, one pass
__global__ __launch_bounds__(256) void k_ealpha_edges(
    const float* __restrict__ ea, const float* __restrict__ wae,
    float* __restrict__ ealpha) {
  __shared__ float w[LL * DD];
  for (int i = threadIdx.x; i < LL * DD; i += 256) w[i] = wae[i];
  __syncthreads();
  int e = blockIdx.x * 8 + (threadIdx.x >> 5);
  if (e >= EE) return;
  int lane = threadIdx.x & 31;
  float4 v = ((const float4*)(ea + (size_t)e * DD))[lane];
#pragma unroll
  for (int l = 0; l < LL; ++l) {
    float4 wv = ((const float4*)(w + l * DD))[lane];
    float p = v.x * wv.x + v.y * wv.y + v.z * wv.z + v.w * wv.w;
    p = waveReduceSum(p);
    if (lane == 0) ealpha[(size_t)l * ETOT + e] = p;
  }
}

// ealpha[l][E+i] = dot(esum[i]/max(cnt[i],1), w_ae[l])   (self-loop edges)
__global__ __launch_bounds__(256) void k_ealpha_loops(
    const float* __restrict__ esum, const float* __restrict__ cnt,
    const float* __restrict__ wae, float* __restrict__ ealpha) {
  __shared__ float w[LL * DD];
  for (int i = threadIdx.x; i < LL * DD; i += 256) w[i] = wae[i];
  __syncthreads();
  int n = blockIdx.x * 8 + (threadIdx.x >> 5);
  if (n >= NN) return;
  int lane = threadIdx.x & 31;
  float inv = 1.0f / fmaxf(cnt[n], 1.0f);
  float4 v = ((const float4*)(esum + (size_t)n * DD))[lane];
  v.x *= inv; v.y *= inv; v.z *= inv; v.w *= inv;
#pragma unroll
  for (int l = 0; l < LL; ++l) {
    float4 wv = ((const float4*)(w + l * DD))[lane];
    float p = v.x * wv.x + v.y * wv.y + v.z * wv.z + v.w * wv.w;
    p = waveReduceSum(p);
    if (lane == 0) ealpha[(size_t)l * ETOT + EE + n] = p;
  }
}

// --------------------------- per-layer kernels -----------------------------

// h = X @ W  (fp32 WMMA 16x16x4). One wave -> one 16x16 tile; 8 waves/block
// share a 16x128 X tile staged in LDS (stride 132 avoids bank conflicts).
// NN % 16 == 0 and no divergence, so EXEC is all-ones at every WMMA.
__global__ __launch_bounds__(256) void k_gemm_h(
    const float* __restrict__ X, const float* __restrict__ W,
    float* __restrict__ H) {
  __shared__ float xt[16][DD + 4];
  const int m0 = blockIdx.x * 16;
  const int t = threadIdx.x;
  for (int i = t; i < 16 * DD; i += 256) {
    int r = i >> 7, c = i & (DD - 1);
    xt[r][c] = X[(size_t)(m0 + r) * DD + c];
  }
  __syncthreads();
  const int wave = t >> 5;
  const int lane = t & 31;
  const int n0 = wave * 16;
  const int lr = lane & 15;  // position within half-wave
  const int hi = lane >> 4;  // 0: K even-pair / M 0..7 side, 1: other half
  v8f acc = {0.f, 0.f, 0.f, 0.f, 0.f, 0.f, 0.f, 0.f};
#pragma unroll 4
  for (int k = 0; k < DD; k += 4) {
    // A 16x4 fp32 frag: v0 = {K=0 | K=2}, v1 = {K=1 | K=3} across lane halves
    v2f a, b;
    a[0] = xt[lr][k + 2 * hi + 0];
    a[1] = xt[lr][k + 2 * hi + 1];
    // B 4x16 fp32 frag: rows striped across lanes within a VGPR (mirror of A)
    b[0] = W[(size_t)(k + 2 * hi + 0) * DD + n0 + lr];
    b[1] = W[(size_t)(k + 2 * hi + 1) * DD + n0 + lr];
    acc = __builtin_amdgcn_wmma_f32_16x16x4_f32(false, a, false, b,
                                                (short)0, acc, false, false);
  }
  // C/D layout: VGPR r, lanes 0-15 -> M=r, lanes 16-31 -> M=8+r
#pragma unroll
  for (int r = 0; r < 8; ++r)
    H[(size_t)(m0 + r + 8 * hi) * DD + n0 + lr] = acc[r];
}

// hs = h@a_s, hd = h@a_d per node; also init segment max / denom
__global__ __launch_bounds__(256) void k_node_dots(
    const float* __restrict__ h, const float* __restrict__ as_,
    const float* __restrict__ ad_, float* __restrict__ hs,
    float* __restrict__ hd, float* __restrict__ m, float* __restrict__ den) {
  int i = blockIdx.x * 8 + (threadIdx.x >> 5);
  if (i >= NN) return;
  int lane = threadIdx.x & 31;
  float4 hv = ((const float4*)(h + (size_t)i * DD))[lane];
  float4 sv = ((const float4*)as_)[lane];
  float4 dv = ((const float4*)ad_)[lane];
  float ps = hv.x * sv.x + hv.y * sv.y + hv.z * sv.z + hv.w * sv.w;
  float pd = hv.x * dv.x + hv.y * dv.y + hv.z * dv.z + hv.w * dv.w;
  ps = waveReduceSum(ps);
  pd = waveReduceSum(pd);
  if (lane == 0) {
    hs[i] = ps; hd[i] = pd;
    m[i] = -__builtin_inff();
    den[i] = 0.0f;
  }
}

// logit = leaky_relu(hs[src]+hd[dst]+ealpha); segment max via CAS atomic
__global__ __launch_bounds__(256) void k_logit_max(
    const int* __restrict__ src, const int* __restrict__ dst,
    const float* __restrict__ hs, const float* __restrict__ hd,
    const float* __restrict__ ealpha, float* __restrict__ logit,
    float* __restrict__ m) {
  int e = blockIdx.x * blockDim.x + threadIdx.x;
  if (e >= ETOT) return;
  int s, d;
  if (e < EE) { s = src[e]; d = dst[e]; } else { s = d = e - EE; }
  float v = hs[s] + hd[d] + ealpha[e];
  v = (v > 0.0f) ? v : NEG_SLOPE * v;
  logit[e] = v;
  atomicMaxF(m + d, v);
}

// ex = exp(logit - m[dst]); denom[dst] += ex
__global__ __launch_bounds__(256) void k_exp_denom(
    const int* __restrict__ dst, const float* __restrict__ logit,
    const float* __restrict__ m, float* __restrict__ ex,
    float* __restrict__ den) {
  int e = blockIdx.x * blockDim.x + threadIdx.x;
  if (e >= ETOT) return;
  int d = (e < EE) ? dst[e] : (e - EE);
  float v = expf(logit[e] - m[d]);
  ex[e] = v;
  atomicAdd(den + d, v);
}

// agg[dst] += h[src] * (ex / (denom[dst] + 1e-16))
__global__ __launch_bounds__(256) void k_scatter(
    const int* __restrict__ src, const int* __restrict__ dst,
    const float* __restrict__ h, const float* __restrict__ ex,
    const float* __restrict__ den, float* __restrict__ agg) {
  int e = blockIdx.x * 8 + (threadIdx.x >> 5);
  if (e >= ETOT) return;
  int lane = threadIdx.x & 31;
  int s, d;
  if (e < EE) { s = src[e]; d = dst[e]; } else { s = d = e - EE; }
  float w = ex[e] / (den[d] + 1e-16f);
  float4 hv = ((const float4*)(h + (size_t)s * DD))[lane];
  float* p = agg + (size_t)d * DD + lane * 4;
  atomicAdd(p + 0, hv.x * w);
  atomicAdd(p + 1, hv.y * w);
  atomicAdd(p + 2, hv.z * w);
  atomicAdd(p + 3, hv.w * w);
}

// x_out = LayerNorm(agg + bias) * gamma + beta   (wave per node)
__global__ __launch_bounds__(256) void k_ln(
    const float* __restrict__ agg, const float* __restrict__ b,
    const float* __restrict__ g, const float* __restrict__ bt,
    float* __restrict__ xout) {
  int i = blockIdx.x * 8 + (threadIdx.x >> 5);
  if (i >= NN) return;
  int lane = threadIdx.x & 31;
  float4 v = ((const float4*)(agg + (size_t)i * DD))[lane];
  float4 bv = ((const float4*)b)[lane];
  v.x += bv.x; v.y += bv.y; v.z += bv.z; v.w += bv.w;
  float mu = waveReduceSum(v.x + v.y + v.z + v.w) * (1.0f / DD);
  float dx = v.x - mu, dy = v.y - mu, dz = v.z - mu, dw = v.w - mu;
  float var = waveReduceSum(dx * dx + dy * dy + dz * dz + dw * dw) * (1.0f / DD);
  float inv = rsqrtf(var + LN_EPS);
  float4 gv = ((const float4*)g)[lane];
  float4 tv = ((const float4*)bt)[lane];
  float4 o;
  o.x = dx * inv * gv.x + tv.x;
  o.y = dy * inv * gv.y + tv.y;
  o.z = dz * inv * gv.z + tv.z;
  o.w = dw * inv * gv.w + tv.w;
  ((float4*)(xout + (size_t)i * DD))[lane] = o;
}

// ------------------------------ launcher -----------------------------------

extern "C" void kernel_launch(void* const* d_in, const int* in_sizes, int n_in,
                              void* d_out, int out_size, void* d_ws,
                              size_t ws_size, hipStream_t stream) {
  const float* x    = (const float*)d_in[0];
  const int*   eidx = (const int*)d_in[1];   // jax x64-off -> int32
  const float* ea   = (const float*)d_in[2];
  const float* W    = (const float*)d_in[3];
  const float* We   = (const float*)d_in[4];
  const float* as_  = (const float*)d_in[5];
  const float* ad_  = (const float*)d_in[6];
  const float* ae   = (const float*)d_in[7];
  const float* bias = (const float*)d_in[8];
  const float* gam  = (const float*)d_in[9];
  const float* bet  = (const float*)d_in[10];
  float* out = (float*)d_out;
  const int* srcv = eidx;
  const int* dstv = eidx + EE;

  // workspace layout (~94 MB of fp32)
  float* ws    = (float*)d_ws;
  float* f_h   = ws;                              // [N,D]
  float* f_agg = f_h + (size_t)NN * DD;           // [N,D] (esum alias in phase0)
  float* f_xA  = f_agg + (size_t)NN * DD;         // [N,D] ping buffer
  float* f_eal = f_xA + (size_t)NN * DD;          // [L,ETOT]
  float* f_log = f_eal + (size_t)LL * ETOT;       // [ETOT]
  float* f_ex  = f_log + ETOT;                    // [ETOT]
  float* f_cnt = f_ex + ETOT;                     // [N]
  float* f_hs  = f_cnt + NN;                      // [N]
  float* f_hd  = f_hs + NN;                       // [N]
  float* f_m   = f_hd + NN;                       // [N]
  float* f_den = f_m + NN;                        // [N]
  float* f_wae = f_den + NN;                      // [L,D]

  // ---- phase 0: layer-invariant precompute (edge_attr read once) ----
  hipMemsetAsync(f_agg, 0, (size_t)NN * DD * sizeof(float), stream);
  hipMemsetAsync(f_cnt, 0, (size_t)NN * sizeof(float), stream);
  k_edge_accum<<<(EE + 7) / 8, 256, 0, stream>>>(ea, dstv, f_agg, f_cnt);
  k_wae<<<LL * DD, DD, 0, stream>>>(We, ae, f_wae);
  k_ealpha_edges<<<(EE + 7) / 8, 256, 0, stream>>>(ea, f_wae, f_eal);
  k_ealpha_loops<<<(NN + 7) / 8, 256, 0, stream>>>(f_agg, f_cnt, f_wae, f_eal);

  // ---- 4 GAT layers; x ping-pongs in->A->out->A->out ----
  for (int l = 0; l < LL; ++l) {
    const float* xin = (l == 0) ? x : ((l == 2) ? out : f_xA);
    float* xout = (l & 1) ? out : f_xA;

    hipMemsetAsync(f_agg, 0, (size_t)NN * DD * sizeof(float), stream);
    k_gemm_h<<<NN / 16, 256, 0, stream>>>(xin, W + (size_t)l * DD * DD, f_h);
    k_node_dots<<<(NN + 7) / 8, 256, 0, stream>>>(
        f_h, as_ + l * DD, ad_ + l * DD, f_hs, f_hd, f_m, f_den);
    k_logit_max<<<(ETOT + 255) / 256, 256, 0, stream>>>(
        srcv, dstv, f_hs, f_hd, f_eal + (size_t)l * ETOT, f_log, f_m);
    k_exp_denom<<<(ETOT + 255) / 256, 256, 0, stream>>>(
        dstv, f_log, f_m, f_ex, f_den);
    k_scatter<<<(ETOT + 7) / 8, 256, 0, stream>>>(
        srcv, dstv, f_h, f_ex, f_den, f_agg);
    k_ln<<<(NN + 7) / 8, 256, 0, stream>>>(
        f_agg, bias + l * DD, gam + l * DD, bet + l * DD, xout);
  }
}